// PointPillarsPostProcess_83021717832095
// MI455X (gfx1250) — compile-verified
//
#include <hip/hip_runtime.h>
#include <hip/hip_bf16.h>
#include <stdint.h>
#include <math.h>

// ---------------- problem constants ----------------
#define B_        4
#define N_        321408
#define KPRE      1000
#define KPOST     300
#define NBINS     4096
#define CAP       4096          // candidate cap per batch (>= ~1000 + one bin)
#define SORT_N    4096
#define TILE      128           // 128 * 2511 == 321408 exactly
#define NTILES    (N_ / TILE)
#define SCORE_THR 0.05f
#define IOU_THR   0.5f
#define PI_F      3.14159265358979323846f

// ---------------- workspace layout (bytes) ----------------
#define HIST_OFF  0u            // B*NBINS u32           = 65536
#define CUT_OFF   65536u        // B u32                 = 16
#define CNT_OFF   65552u        // B u32                 = 16
#define CAND_OFF  65568u        // B*CAP u64             = 131072
#define TOPS_OFF  196640u       // B*KPRE f32            = 16000
#define TOPI_OFF  212640u       // B*KPRE i32            = 16000  -> total 228640

// ---------------- CDNA5 async global->LDS helpers (ASYNCcnt path) ----------------
__device__ __forceinline__ void async_ld_b128(unsigned lds_off, const void* g) {
  // gfx1250: copies 16B per lane from global vaddr to LDS offset vdst; tracked with ASYNCcnt
  asm volatile("global_load_async_to_lds_b128 %0, %1, off" :: "v"(lds_off), "v"(g) : "memory");
}
__device__ __forceinline__ void async_wait0() {
  asm volatile("s_wait_asynccnt 0x0" ::: "memory");
}

// ---------------- kernels ----------------
__global__ void pp_zero(unsigned* __restrict__ p, int n) {
  for (int i = blockIdx.x * blockDim.x + threadIdx.x; i < n; i += gridDim.x * blockDim.x)
    p[i] = 0u;
}

// Stage a 128-anchor tile of cls_preds (1536 B) into LDS with async b128 loads,
// then return this thread's sigmoid(max logit) score.
__device__ __forceinline__ float tile_score(const float* __restrict__ cls,
                                            float* __restrict__ sh, int b, int tile, int t) {
  size_t base = ((size_t)b * N_ + (size_t)tile * TILE) * 3;
  const float* g = cls + base;
  unsigned l0 = (unsigned)(size_t)(&sh[0]);               // low 32 bits = LDS offset
  if (t < (TILE * 3) / 4) {                               // 96 lanes x 16B = whole tile
    async_ld_b128(l0 + (unsigned)t * 16u, g + t * 4);
  }
  async_wait0();
  __syncthreads();
  float c0 = sh[3 * t + 0], c1 = sh[3 * t + 1], c2 = sh[3 * t + 2];
  float m = fmaxf(c0, fmaxf(c1, c2));
  return 1.0f / (1.0f + expf(-m));                        // max(sigmoid) == sigmoid(max)
}

__global__ void pp_hist(const float* __restrict__ cls, unsigned* __restrict__ hist) {
  __shared__ float sh[TILE * 3];
  int blk = blockIdx.x, b = blk / NTILES, tile = blk - b * NTILES, t = threadIdx.x;
  float s = tile_score(cls, sh, b, tile, t);
  if (s >= SCORE_THR) {
    int bin = (int)(s * (float)NBINS);
    bin = bin < 0 ? 0 : (bin > NBINS - 1 ? NBINS - 1 : bin);
    atomicAdd(&hist[b * NBINS + bin], 1u);
  }
}

// Parallel suffix-sum over the histogram; cutoff = largest bin with suffix >= KPRE.
__global__ __launch_bounds__(1024) void pp_cutoff(const unsigned* __restrict__ hist,
                                                  unsigned* __restrict__ cut) {
  __shared__ unsigned s[NBINS];                            // 16 KB
  int b = blockIdx.x, tid = threadIdx.x;
  for (int i = tid; i < NBINS; i += 1024) s[i] = hist[b * NBINS + i];
  __syncthreads();
  for (int off = 1; off < NBINS; off <<= 1) {              // Hillis-Steele suffix scan
    unsigned v[NBINS / 1024];
    int r = 0;
    for (int i = tid; i < NBINS; i += 1024, ++r)
      v[r] = s[i] + ((i + off < NBINS) ? s[i + off] : 0u);
    __syncthreads();
    r = 0;
    for (int i = tid; i < NBINS; i += 1024, ++r) s[i] = v[r];
    __syncthreads();
  }
  for (int i = tid; i < NBINS; i += 1024) {
    bool hit = (s[i] >= (unsigned)KPRE) && (i == NBINS - 1 || s[i + 1] < (unsigned)KPRE);
    if (hit) cut[b] = (unsigned)i;                         // unique by monotonicity
  }                                                        // none -> stays 0 (pp_zero)
}

__global__ void pp_compact(const float* __restrict__ cls, const unsigned* __restrict__ cut,
                           unsigned* __restrict__ cnt, unsigned long long* __restrict__ cand) {
  __shared__ float sh[TILE * 3];
  int blk = blockIdx.x, b = blk / NTILES, tile = blk - b * NTILES, t = threadIdx.x;
  float s = tile_score(cls, sh, b, tile, t);
  if (s >= SCORE_THR) {
    int bin = (int)(s * (float)NBINS);
    bin = bin < 0 ? 0 : (bin > NBINS - 1 ? NBINS - 1 : bin);
    if (bin >= (int)cut[b]) {
      unsigned pos = atomicAdd(&cnt[b], 1u);
      if (pos < (unsigned)CAP) {
        unsigned n  = (unsigned)(tile * TILE + t);
        unsigned sb = __float_as_uint(s);
        // descending score, ascending index tie-break, via one descending u64 sort
        cand[(size_t)b * CAP + pos] = (((unsigned long long)sb) << 32) | (unsigned)(~n);
      }
    }
  }
}

__global__ __launch_bounds__(1024) void pp_sort(const unsigned long long* __restrict__ cand,
                                                const unsigned* __restrict__ cnt,
                                                float* __restrict__ topS, int* __restrict__ topI) {
  __shared__ unsigned long long keys[SORT_N];              // 32 KB
  int b = blockIdx.x, tid = threadIdx.x, nth = blockDim.x;
  unsigned nc = cnt[b]; if (nc > (unsigned)CAP) nc = CAP;
  for (int i = tid; i < SORT_N; i += nth)
    keys[i] = (i < (int)nc) ? cand[(size_t)b * CAP + i] : 0ull;
  __syncthreads();
  for (int k = 2; k <= SORT_N; k <<= 1) {
    for (int j = k >> 1; j > 0; j >>= 1) {
      for (int i = tid; i < SORT_N; i += nth) {
        int ixj = i ^ j;
        if (ixj > i) {
          unsigned long long a = keys[i], c = keys[ixj];
          bool up = ((i & k) == 0);
          if (up ? (a < c) : (a > c)) { keys[i] = c; keys[ixj] = a; }  // descending
        }
      }
      __syncthreads();
    }
  }
  for (int i = tid; i < KPRE; i += nth) {
    unsigned long long kk = keys[i];
    topS[b * KPRE + i] = __uint_as_float((unsigned)(kk >> 32)); // pad -> 0.0f
    topI[b * KPRE + i] = (int)(~((unsigned)kk));                // pad -> -1
  }
}

// Bitmask NMS, entirely in (large, CDNA5-only) dynamic LDS: ~185 KB of the 320 KB/WGP.
// Phase 1: decode top-1000. Phase 2: 1000x1000 suppression bit-matrix (1 IoU/lane,
// __ballot per wave32 -> mask word). Phase 3: ONE wave does the sequential greedy
// scan (shfl + LDS word per box, zero block barriers). Phase 4/5: select + emit.
extern __shared__ unsigned pp_smem[];

__global__ __launch_bounds__(1024) void pp_nms(
    const float* __restrict__ boxp, const float* __restrict__ cls,
    const float* __restrict__ dirp, const float* __restrict__ anch,
    const float* __restrict__ topS, const int* __restrict__ topI,
    float* __restrict__ out) {
  unsigned* base = pp_smem;
  float* bxs = (float*)(base + 0 * KPRE);
  float* bys = (float*)(base + 1 * KPRE);
  float* bzs = (float*)(base + 2 * KPRE);
  float* bws = (float*)(base + 3 * KPRE);
  float* bls = (float*)(base + 4 * KPRE);
  float* bhs = (float*)(base + 5 * KPRE);
  float* brs = (float*)(base + 6 * KPRE);
  float* x1s = (float*)(base + 7 * KPRE);
  float* x2s = (float*)(base + 8 * KPRE);
  float* y1s = (float*)(base + 9 * KPRE);
  float* y2s = (float*)(base + 10 * KPRE);
  float* ars = (float*)(base + 11 * KPRE);
  float* scs = (float*)(base + 12 * KPRE);
  int*   gidx = (int*)(base + 13 * KPRE);
  int*   vld  = (int*)(base + 14 * KPRE);
  unsigned* mask  = base + 15 * KPRE;                      // KPRE*32 words = 128 KB
  unsigned* keepw = base + 15 * KPRE + KPRE * 32;          // 32 words
  int*      sel   = (int*)(keepw + 32);                    // 300 words

  int b = blockIdx.x, tid = threadIdx.x, nth = blockDim.x;
  int wid = tid >> 5, lane = tid & 31;

  // ---- phase 1: gather + decode only the top-1000 survivors ----
  for (int i = tid; i < KPRE; i += nth) {
    float sc = topS[b * KPRE + i];
    int idx  = topI[b * KPRE + i];
    scs[i] = sc; gidx[i] = idx;
    bool valid = (sc >= SCORE_THR) && (idx >= 0) && (idx < N_);
    float xg=0,yg=0,zg=0,wg=0,lg=0,hg=0,rg=0,x1=0,x2=0,y1=0,y2=0,ar=0;
    if (valid) {
      const float* a = anch + (size_t)idx * 7;
      const float* d = boxp + ((size_t)b * N_ + (size_t)idx) * 7;
      float xa=a[0], ya=a[1], za=a[2], wa=a[3], la=a[4], ha=a[5], ra=a[6];
      float xt=d[0], yt=d[1], zt=d[2], wt=d[3], lt=d[4], ht=d[5], rt=d[6];
      za += ha * 0.5f;
      float diag = sqrtf(la * la + wa * wa);
      xg = xt * diag + xa; yg = yt * diag + ya; zg = zt * ha + za;
      wg = expf(wt) * wa;  lg = expf(lt) * la;  hg = expf(ht) * ha;
      zg -= hg * 0.5f;     rg = rt + ra;
      float cc = fabsf(cosf(rg)), ss = fabsf(sinf(rg));
      float hx = 0.5f * (wg * cc + lg * ss), hy = 0.5f * (wg * ss + lg * cc);
      x1 = xg - hx; x2 = xg + hx; y1 = yg - hy; y2 = yg + hy;
      ar = (x2 - x1) * (y2 - y1);
    }
    bxs[i]=xg; bys[i]=yg; bzs[i]=zg; bws[i]=wg; bls[i]=lg; bhs[i]=hg; brs[i]=rg;
    x1s[i]=x1; x2s[i]=x2; y1s[i]=y1; y2s[i]=y2; ars[i]=ar;
    vld[i] = valid ? 1 : 0;
  }
  __syncthreads();

  // ---- phase 2: suppression bit-matrix; wave task = (row i, word w) ----
  for (int task = wid; task < KPRE * 32; task += 32) {
    int i = task >> 5, w = task & 31;
    int j = (w << 5) | lane;
    bool sup = false;
    if (j < KPRE && j > i) {
      float ix = fmaxf(fminf(x2s[i], x2s[j]) - fmaxf(x1s[i], x1s[j]), 0.0f);
      float iy = fmaxf(fminf(y2s[i], y2s[j]) - fmaxf(y1s[i], y1s[j]), 0.0f);
      float inter = ix * iy;
      float un = ars[i] + ars[j] - inter;
      float iou = (un > 0.0f) ? inter / fmaxf(un, 1e-12f) : 0.0f;
      sup = iou > IOU_THR;
    }
    unsigned m = (unsigned)__ballot(sup);                  // wave32 ballot -> 32 bits
    if (lane == 0) mask[(i << 5) + w] = m;
  }
  __syncthreads();

  // ---- phase 3: sequential greedy scan, single wave, no block barriers ----
  if (wid == 0) {
    unsigned kw = 0;                                       // lane owns word `lane`
    for (int bit = 0; bit < 32; ++bit) {
      int j = (lane << 5) | bit;
      if (j < KPRE && vld[j]) kw |= (1u << bit);
    }
    for (int i = 0; i < KPRE; ++i) {
      unsigned aliveW = __shfl(kw, i >> 5, 32);
      if ((aliveW >> (i & 31)) & 1u)
        kw &= ~mask[(i << 5) + lane];
    }
    keepw[lane] = kw;
  }
  __syncthreads();

  // ---- phase 4: sel = kept (score order) then non-kept (ascending index) ----
  if (tid == 0) {
    int p = 0;
    for (int i = 0; i < KPRE && p < KPOST; ++i)
      if ((keepw[i >> 5] >> (i & 31)) & 1u) sel[p++] = i;
    for (int i = 0; i < KPRE && p < KPOST; ++i)
      if (!((keepw[i >> 5] >> (i & 31)) & 1u)) sel[p++] = i;
  }
  __syncthreads();

  // ---- phase 5: direction fix, range filter, masked outputs ----
  for (int p = tid; p < KPOST; p += nth) {
    int i = sel[p];
    int kflag = (int)((keepw[i >> 5] >> (i & 31)) & 1u);
    int aidx  = gidx[i];
    float x=bxs[i], y=bys[i], z=bzs[i], w=bws[i], l=bls[i], h=bhs[i], r=brs[i], sc=scs[i];
    int label = 0, dirlab = 0;
    if (aidx >= 0) {
      const float* cp = cls + ((size_t)b * N_ + (size_t)aidx) * 3;
      float c0 = cp[0], c1 = cp[1], c2 = cp[2];
      float best = c0;
      if (c1 > best) { best = c1; label = 1; }
      if (c2 > best) { best = c2; label = 2; }
      const float* dp = dirp + ((size_t)b * N_ + (size_t)aidx) * 2;
      dirlab = (dp[1] > dp[0]) ? 1 : 0;
    }
    bool opp = ((r > 0.0f) != (dirlab == 1));              // DIR_OFFSET == 0
    float rf = r + (opp ? PI_F : 0.0f);
    bool inr = (x >= 0.0f) && (x <= 69.12f) &&
               (y >= -39.68f) && (y <= 39.68f) &&
               (z >= -5.0f)  && (z <= 5.0f);
    bool vout = (kflag != 0) && inr;
    float ob[7] = {x, y, z, w, l, h, rf};
    size_t bo = ((size_t)b * KPOST + p) * 7;
    #pragma unroll
    for (int c = 0; c < 7; ++c) out[bo + c] = vout ? ob[c] : 0.0f;
    size_t so = (size_t)b * KPOST + p;
    out[8400  + so] = vout ? (float)label : 0.0f;          // labels
    out[9600  + so] = vout ? sc : 0.0f;                    // scores
    out[10800 + so] = vout ? 1.0f : 0.0f;                  // valid_out
  }
}

// ---------------- launch ----------------
extern "C" void kernel_launch(void* const* d_in, const int* in_sizes, int n_in,
                              void* d_out, int out_size, void* d_ws, size_t ws_size,
                              hipStream_t stream) {
  const float* boxp = (const float*)d_in[0];
  const float* cls  = (const float*)d_in[1];
  const float* dirp = (const float*)d_in[2];
  const float* anch = (const float*)d_in[3];
  float* out = (float*)d_out;
  char*  ws  = (char*)d_ws;

  unsigned* hist = (unsigned*)(ws + HIST_OFF);
  unsigned* cut  = (unsigned*)(ws + CUT_OFF);
  unsigned* cnt  = (unsigned*)(ws + CNT_OFF);
  unsigned long long* cand = (unsigned long long*)(ws + CAND_OFF);
  float* topS = (float*)(ws + TOPS_OFF);
  int*   topI = (int*)(ws + TOPI_OFF);

  // dynamic LDS for pp_nms: 15*KPRE + KPRE*32 + 32 + KPOST words (~185 KB; CDNA5 has 320 KB/WGP)
  const unsigned nmsSmem = (15u * KPRE + 32u * KPRE + 32u + (unsigned)KPOST) * 4u;

  pp_zero   <<<64, 256, 0, stream>>>((unsigned*)ws, B_ * NBINS + 2 * B_);
  pp_hist   <<<B_ * NTILES, TILE, 0, stream>>>(cls, hist);
  pp_cutoff <<<B_, 1024, 0, stream>>>(hist, cut);
  pp_compact<<<B_ * NTILES, TILE, 0, stream>>>(cls, cut, cnt, cand);
  pp_sort   <<<B_, 1024, 0, stream>>>(cand, cnt, topS, topI);
  pp_nms    <<<B_, 1024, nmsSmem, stream>>>(boxp, cls, dirp, anch, topS, topI, out);
}